// FrustumRangePreprocessor_63342177681969
// MI455X (gfx1250) — compile-verified
//
#include <hip/hip_runtime.h>
#include <hip/hip_bf16.h>

// FrustumRangePreprocessor for MI455X (gfx1250).
// Memory-bound op (~210 MB traffic -> ~9 us at 23.3 TB/s), zero matrix FLOPs,
// so no WMMA by design. gfx1250 paths used:
//   - global_load_b128 / global_store_b128 vector memory ops
//   - global_atomic_add_u32 on a 40 MiB L2-resident packed-u16 histogram
//   - TH=NT cache hints (__builtin_nontemporal_*) on all streaming data
//   - global_prefetch_b8 (__builtin_prefetch)
//   - ASYNC path: global_load_async_to_lds_b128 + s_wait_asynccnt +
//     ds_load_b128, double-buffered, for the 40 MB histogram read-back

typedef float        v4f __attribute__((ext_vector_type(4)));
typedef unsigned int v4u __attribute__((ext_vector_type(4)));

// Builtin-exact types for the async copy (signature from clang diagnostic:
// "vV4i*1 V4i*3 Ii Ii" -> v4i addrspace(1)*, v4i addrspace(3)*, imm, imm).
typedef int v4i_vec __attribute__((__vector_size__(16)));
typedef __attribute__((address_space(1))) v4i_vec* as1_v4i_p;
typedef __attribute__((address_space(3))) v4i_vec* as3_v4i_p;

#if defined(__has_builtin)
#if __has_builtin(__builtin_amdgcn_global_load_async_to_lds_b128) && \
    __has_builtin(__builtin_amdgcn_s_wait_asynccnt)
#define FRUSTUM_HAVE_ASYNC_LDS 1
#endif
#endif

namespace {
constexpr int   kH = 64;
constexpr int   kW = 2048;
constexpr int   kC = 20;               // NUM_CLASSES
constexpr int   kWpc = kC / 2;         // u32 words per cell (2 u16 counters/word)
constexpr int   kB = 8;
constexpr int   kIgnore = 255;
constexpr int   kTpb = 256;            // 8 wave32's per block
constexpr float kAbsFovDown = 0.43633231299858238f;           // |deg2rad(-25)|
constexpr float kInvFov     = 1.0f / 0.48869219055841229f;    // 1/(FOV_UP+|FOV_DOWN|)
constexpr float kInvPi      = 0.31830988618379067f;
}

// ---------------------------------------------------------------- zero counts
// Default (RT) stores: we WANT these lines resident in L2 for the atomics.
__global__ void frustum_zero_counts(v4u* __restrict__ p, int n16) {
  int i = blockIdx.x * blockDim.x + threadIdx.x;
  int stride = gridDim.x * blockDim.x;
  const v4u z = {0u, 0u, 0u, 0u};
  for (; i < n16; i += stride) {
    p[i] = z;                                       // global_store_b128
  }
}

// ------------------------------------------------ project + scatter histogram
__global__ void frustum_points_kernel(const v4f*    __restrict__ pts,
                                      const int*    __restrict__ labels,
                                      float*        __restrict__ out_pts,    // [B*N,4]
                                      float*        __restrict__ out_coors,  // [B*N,3]
                                      float*        __restrict__ out_mask,   // [B*N]
                                      unsigned int* __restrict__ counts,     // [cells*10]
                                      int total, int n_per_batch) {
  int i = blockIdx.x * blockDim.x + threadIdx.x;
  int stride = gridDim.x * blockDim.x;
  for (; i < total; i += stride) {
    if (i + stride < total) {
      __builtin_prefetch(&pts[i + stride], 0, 1);   // global_prefetch_b8
    }
    v4f p = __builtin_nontemporal_load(pts + i);    // global_load_b128 (TH_NT)
    float x = p.x, y = p.y, z = p.z;

    float depth = fmaxf(sqrtf(x * x + y * y + z * z), 1e-6f);
    float yaw   = -atan2f(y, x);
    float s     = fminf(fmaxf(z / depth, -1.0f), 1.0f);
    float pitch = asinf(s);

    int px = (int)floorf(0.5f * (yaw * kInvPi + 1.0f) * (float)kW);
    px = min(max(px, 0), kW - 1);
    int py = (int)floorf((1.0f - (pitch + kAbsFovDown) * kInvFov) * (float)kH);
    py = min(max(py, 0), kH - 1);
    int b = i / n_per_batch;

    __builtin_nontemporal_store(p, (v4f*)out_pts + i);   // global_store_b128 (TH_NT)

    __builtin_nontemporal_store((float)b,  out_coors + 3 * i + 0);
    __builtin_nontemporal_store((float)py, out_coors + 3 * i + 1);
    __builtin_nontemporal_store((float)px, out_coors + 3 * i + 2);

    int lab = labels[i];
    __builtin_nontemporal_store((float)lab, out_mask + i);

    if (lab != kIgnore) {
      // Packed u16 histogram: 2 classes per u32 word. Hot clamp rows peak at
      // ~69 pts/cell over 20 classes; 16-bit ceiling 65535 -> no carry, exact.
      int cell = (b * kH + py) * kW + px;
      unsigned word = (unsigned)cell * kWpc + (unsigned)(lab >> 1);
      unsigned add  = (lab & 1) ? 0x10000u : 1u;
      atomicAdd(&counts[word], add);                // global_atomic_add_u32
    }
  }
}

// --------------------------------------- majority vote for one 2-cell segment
// ww = 20 u32 words: words [0,10) = cell A, [10,20) = cell B (u16-packed).
__device__ __forceinline__ void frustum_vote2(const unsigned* ww, float* seg, int pair) {
#pragma unroll
  for (int half = 0; half < 2; ++half) {
    const unsigned* w = ww + half * kWpc;
    unsigned best = w[0] & 0xffffu;                 // class 0
    unsigned tot  = best;
    int bi = 0;
#pragma unroll
    for (int c = 1; c < kC; ++c) {
      unsigned v = (c & 1) ? (w[c >> 1] >> 16) : (w[c >> 1] & 0xffffu);
      tot += v;
      if (v > best) { best = v; bi = c; }           // strict '>' == first-max tie-break
    }
    float r = (tot > 0u) ? (float)bi : (float)kIgnore;
    __builtin_nontemporal_store(r, seg + 2 * pair + half);
  }
}

#if defined(FRUSTUM_HAVE_ASYNC_LDS)
// ---------------------- per-cell majority vote, async-to-LDS staged pipeline
// One thread owns TWO cells per tile: 80 B = 5 B128 async copies into LDS,
// double-buffered across grid-stride tiles, consumed with ds_load_b128.
__global__ void frustum_argmax_kernel(const unsigned int* __restrict__ counts,
                                      float* __restrict__ seg, int npairs) {
  __shared__ unsigned int lds[2][kTpb * 2 * kWpc]; // 2 x 20 KiB = 40 KiB
  const int tid = threadIdx.x;
  const int stride = gridDim.x * blockDim.x;
  int t = blockIdx.x * blockDim.x + tid;

  auto issue = [&](int buf, int pair) {
    const unsigned int* gsrc = counts + (size_t)pair * (2 * kWpc);
    unsigned int* ldst = &lds[buf][tid * (2 * kWpc)];
#pragma unroll
    for (int k = 0; k < 5; ++k) {
      __builtin_amdgcn_global_load_async_to_lds_b128(
          (as1_v4i_p)(gsrc + 4 * k),                // addrspace(1) v4i*
          (as3_v4i_p)(ldst + 4 * k),                // addrspace(3) v4i* (LDS offset)
          0, 0);                                    // global_load_async_to_lds_b128
    }
  };

  if (t < npairs) issue(0, t);
  int buf = 0;
  for (; t < npairs; t += stride) {
    const int nxt = t + stride;
    if (nxt < npairs) {
      issue(buf ^ 1, nxt);                          // fill other buffer
      __builtin_amdgcn_s_wait_asynccnt(5);          // loads retire in order:
                                                    // <=5 outstanding => tile t done
    } else {
      __builtin_amdgcn_s_wait_asynccnt(0);
    }
    // Consume own 80 B segment: 5 aligned ds_load_b128.
    const v4u* l4 = (const v4u*)&lds[buf][tid * (2 * kWpc)];
    unsigned w[2 * kWpc];
#pragma unroll
    for (int k = 0; k < 5; ++k) {
      v4u q = l4[k];                                // ds_load_b128
      w[4 * k + 0] = q.x; w[4 * k + 1] = q.y;
      w[4 * k + 2] = q.z; w[4 * k + 3] = q.w;
    }
    frustum_vote2(w, seg, t);
    buf ^= 1;
  }
}
#else
// ------------------------------- fallback: direct B128 reads (still 2 cells)
__global__ void frustum_argmax_kernel(const unsigned int* __restrict__ counts,
                                      float* __restrict__ seg, int npairs) {
  int t = blockIdx.x * blockDim.x + threadIdx.x;
  int stride = gridDim.x * blockDim.x;
  for (; t < npairs; t += stride) {
    const v4u* c4 = (const v4u*)(counts + (size_t)t * (2 * kWpc));
    unsigned w[2 * kWpc];
#pragma unroll
    for (int k = 0; k < 5; ++k) {
      v4u q = __builtin_nontemporal_load(c4 + k);   // global_load_b128 (TH_NT)
      w[4 * k + 0] = q.x; w[4 * k + 1] = q.y;
      w[4 * k + 2] = q.z; w[4 * k + 3] = q.w;
    }
    frustum_vote2(w, seg, t);
  }
}
#endif

// ------------------------------------------------------------------- launcher
extern "C" void kernel_launch(void* const* d_in, const int* in_sizes, int n_in,
                              void* d_out, int out_size, void* d_ws, size_t ws_size,
                              hipStream_t stream) {
  const v4f* pts    = (const v4f*)d_in[0];         // [B, N, 4] fp32
  const int* labels = (const int*)d_in[1];         // [B, N] int

  const int total       = in_sizes[1];             // B*N = 2,400,000
  const int n_per_batch = total / kB;              // N = 300,000
  const int ncells      = kB * kH * kW;            // 1,048,576
  const size_t count_words = (size_t)ncells * kWpc; // 10,485,760 u32 = 40 MiB

  float* out       = (float*)d_out;
  float* out_pts   = out;                                   // [B*N*4]
  float* out_coors = out + (size_t)total * 4;               // [B*N*3]
  float* out_seg   = out + (size_t)total * 7;               // [B*H*W]
  float* out_mask  = out + (size_t)total * 7 + ncells;      // [B*N]

  unsigned int* counts = (unsigned int*)d_ws;      // 40 MiB scratch, L2-resident

  // 1) zero the packed histogram every call (ws not re-poisoned between replays)
  const int n16 = (int)(count_words / 4);          // 2,621,440 v4u stores
  frustum_zero_counts<<<2048, kTpb, 0, stream>>>((v4u*)counts, n16);

  // 2) projection + pass-through outputs + vote scatter
  int pblocks = (total + kTpb - 1) / kTpb;
  if (pblocks > 4096) pblocks = 4096;              // grid-stride, enables prefetch-ahead
  frustum_points_kernel<<<pblocks, kTpb, 0, stream>>>(
      pts, labels, out_pts, out_coors, out_mask, counts, total, n_per_batch);

  // 3) argmax over 20 classes, two cells per thread; 512 blocks -> 4 tiles per
  //    thread so the async double-buffer pipeline actually overlaps.
  const int npairs = ncells / 2;                   // 524,288
  frustum_argmax_kernel<<<512, kTpb, 0, stream>>>(counts, out_seg, npairs);
}